// DynamicFeedForward_23459111371128
// MI455X (gfx1250) — compile-verified
//
#include <hip/hip_runtime.h>
#include <hip/hip_bf16.h>

// DynamicFeedForward: out[b,s,m] = relu( dot(x[b,s,:], W[mask[b,s,m],:]) + bias[mask[b,s,m]] )
//   input_value: [4,2048,512] f32   mask_tensor: [4,2048,32] i64
//   weight:      [50000,512]  f32   bias:        [50000]     f32
//   out:         [4,2048,32]  f32
//
// Strategy (MI455X / gfx1250):
//  - Memory bound: 512MB of gathered weight rows, but weight (102MB) fits in 192MB L2.
//  - One wave32 per token. V_WMMA_F32_16X16X4_F32 with M = 16 candidate rows,
//    B = x broadcast along N. Two accumulator groups cover all 32 candidates.
//  - A-frag: lane L (L<16) loads W[cand[L]][k..k+1], lane L+16 loads W[cand[L]][k+2..k+3]
//    (one b64 per lane per K-chunk; sequential row walk -> full cache-line reuse).
//  - B-frag: lanes 0-15 load x[k..k+1], lanes 16-31 load x[k+2..k+3] (L0 broadcast).
//  - Lanes 0 and 16 (N==0 columns) hold the results; they apply bias + ReLU and store.

typedef __attribute__((ext_vector_type(2))) float v2f;
typedef __attribute__((ext_vector_type(8))) float v8f;

#define HIDDEN 512
#define NCAND 32
#define WAVES_PER_BLOCK 8

__global__ __launch_bounds__(WAVES_PER_BLOCK * 32)
void dynamic_ffn_wmma_f32(const float* __restrict__ input,      // [8192, 512]
                          const long long* __restrict__ mask,   // [8192, 32]
                          const float* __restrict__ weight,     // [50000, 512]
                          const float* __restrict__ bias,       // [50000]
                          float* __restrict__ out)              // [8192, 32]
{
    const int lane  = threadIdx.x & 31;
    const int wave  = threadIdx.x >> 5;
    const int token = blockIdx.x * WAVES_PER_BLOCK + wave;

    const float*     x    = input + (size_t)token * HIDDEN;
    const long long* cand = mask  + (size_t)token * NCAND;

    const int m    = lane & 15;   // candidate slot within group
    const int half = lane >> 4;   // 0: K=0,1  1: K=2,3 (per 16x4 f32 A layout)

    const long long r0 = cand[m];        // group 0 row
    const long long r1 = cand[16 + m];   // group 1 row

    const float* w0 = weight + (size_t)r0 * HIDDEN + 2 * half;
    const float* w1 = weight + (size_t)r1 * HIDDEN + 2 * half;
    const float* xp = x + 2 * half;

    // Pull the gathered rows toward L2/L0 early (global_prefetch_b8).
    __builtin_prefetch(w0, 0, 0);
    __builtin_prefetch(w1, 0, 0);
    __builtin_prefetch(w0 + 32, 0, 0);
    __builtin_prefetch(w1 + 32, 0, 0);

    v8f acc0 = {};
    v8f acc1 = {};

#pragma unroll 8
    for (int k = 0; k < HIDDEN; k += 4) {
        // A fragments: lane<16 -> W[row][k..k+1], lane>=16 -> W[row][k+2..k+3]
        const v2f a0 = *(const v2f*)(w0 + k);
        const v2f a1 = *(const v2f*)(w1 + k);
        // B fragment (x broadcast along N): lanes 0-15 x[k..k+1], lanes 16-31 x[k+2..k+3]
        const v2f b  = *(const v2f*)(xp + k);

        acc0 = __builtin_amdgcn_wmma_f32_16x16x4_f32(
            /*neg_a=*/false, a0, /*neg_b=*/false, b,
            /*c_mod=*/(short)0, acc0, /*reuse_a=*/false, /*reuse_b=*/false);
        acc1 = __builtin_amdgcn_wmma_f32_16x16x4_f32(
            /*neg_a=*/false, a1, /*neg_b=*/false, b,
            /*c_mod=*/(short)0, acc1, /*reuse_a=*/false, /*reuse_b=*/false);
    }

    // D layout (16x16 f32 C/D): VGPR r, lane 0  -> D[M=r][N=0]
    //                           VGPR r, lane 16 -> D[M=8+r][N=0]
    // Lanes 0 and 16 each own 8 results per group.
    if ((lane & 15) == 0) {
        float* o = out + (size_t)token * NCAND;
        const int mbase = half * 8;
#pragma unroll
        for (int r = 0; r < 8; ++r) {
            const long long c0 = cand[mbase + r];
            float v0 = acc0[r] + bias[c0];
            o[mbase + r] = v0 > 0.0f ? v0 : 0.0f;

            const long long c1 = cand[16 + mbase + r];
            float v1 = acc1[r] + bias[c1];
            o[16 + mbase + r] = v1 > 0.0f ? v1 : 0.0f;
        }
    }
}

extern "C" void kernel_launch(void* const* d_in, const int* in_sizes, int n_in,
                              void* d_out, int out_size, void* d_ws, size_t ws_size,
                              hipStream_t stream) {
    const float*     input  = (const float*)d_in[0];      // [4,2048,512] f32
    const long long* mask   = (const long long*)d_in[1];  // [4,2048,32]  i64
    const float*     weight = (const float*)d_in[2];      // [50000,512]  f32
    const float*     bias   = (const float*)d_in[3];      // [50000]      f32
    float*           out    = (float*)d_out;              // [4,2048,32]  f32

    const int tokens = in_sizes[0] / HIDDEN;              // 8192
    const int blocks = tokens / WAVES_PER_BLOCK;          // 1024

    dynamic_ffn_wmma_f32<<<blocks, WAVES_PER_BLOCK * 32, 0, stream>>>(
        input, mask, weight, bias, out);
}